// PNet_30193620091460
// MI455X (gfx1250) — compile-verified
//
#include <hip/hip_runtime.h>
#include <stdint.h>

// out[i, j] = z[i, w[i]] * W[j, w[i]] + b[j]   (N=16384, L=4096, f32)
// Memory-bound (AI ~0.4 FLOP/B) -> no WMMA; optimize data movement.
// Pass 1: Wt = W^T via async-staged LDS tiles (coalesced both sides).
// Pass 2: per-row gather of Wt[w[i],:]: per-wave async global->LDS chunks
//         (ASYNCcnt, no block barrier), FMA with bias, NT B128 stores.
// Fast path is compiled for L=4096 / 256 threads so all index math folds.

typedef float v4f __attribute__((ext_vector_type(4)));

__device__ __forceinline__ void async_b128(uint32_t lds_off, uint64_t gaddr) {
  asm volatile("global_load_async_to_lds_b128 %0, %1, off"
               :: "v"(lds_off), "v"(gaddr) : "memory");
}
__device__ __forceinline__ void wait_async0() {
  asm volatile("s_wait_asynccnt 0" ::: "memory");
}

constexpr int LC      = 4096;           // fixed problem width
constexpr int THREADS = 256;            // 8 wave32
constexpr int NWAVES  = THREADS / 32;
constexpr int CHUNK   = LC / NWAVES;    // 512 floats per wave
constexpr int ITERS   = CHUNK / 128;    // 4 float4-iters per lane

// One block per output row i. Each wave async-stages its own LDS chunk of
// Wt row w[i] and consumes it after its own s_wait_asynccnt (no barrier).
__global__ void __launch_bounds__(THREADS)
PNet_row_kernel(const int* __restrict__ w,
                const float* __restrict__ z,
                const float* __restrict__ Wt,
                const float* __restrict__ b,
                float* __restrict__ out) {
  __shared__ float srow[LC];
  const int i  = blockIdx.x;
  const int wi = w[i];
  const float zw = z[i * LC + wi];
  const float* __restrict__ src = Wt + wi * LC;

  const int base = (threadIdx.x >> 5) * CHUNK + (threadIdx.x & 31) * 4;

#pragma unroll
  for (int k = 0; k < ITERS; ++k) {
    const int off = base + k * 128;
    async_b128((uint32_t)(uintptr_t)(&srow[off]),
               (uint64_t)(uintptr_t)(src + off));
  }
  wait_async0();                        // this wave's chunk is in LDS

  float* __restrict__ orow = out + i * LC;
#pragma unroll
  for (int k = 0; k < ITERS; ++k) {
    const int off = base + k * 128;
    v4f wv = *(const v4f*)(&srow[off]); // ds_load_b128
    v4f bv = *(const v4f*)(&b[off]);    // L2-resident
    v4f o;
    o.x = fmaf(zw, wv.x, bv.x);
    o.y = fmaf(zw, wv.y, bv.y);
    o.z = fmaf(zw, wv.z, bv.z);
    o.w = fmaf(zw, wv.w, bv.w);
    __builtin_nontemporal_store(o, (v4f*)(&orow[off]));  // keep L2 for Wt/b
  }
}

#define TILE 32
#define TPAD 33

// 32x32 tile transpose; tile load goes straight to LDS via async copy.
__global__ void __launch_bounds__(THREADS)
PNet_transpose_kernel(const float* __restrict__ in, float* __restrict__ out) {
  __shared__ float tile[TILE * TPAD];
  const int t  = threadIdx.x;           // 256 threads
  const int r  = t >> 3;                // 0..31 tile row
  const int c  = (t & 7) * 4;           // 0..28 tile col (float4)
  const int bx = blockIdx.x, by = blockIdx.y;

  async_b128((uint32_t)(uintptr_t)(&tile[r * TPAD + c]),
             (uint64_t)(uintptr_t)(in + (by * TILE + r) * LC + bx * TILE + c));
  wait_async0();
  __syncthreads();                      // tile filled by all waves

  v4f o;  // Wt[bx*32+r][by*32+c+q] = W[by*32+c+q][bx*32+r]
  o.x = tile[(c + 0) * TPAD + r];
  o.y = tile[(c + 1) * TPAD + r];
  o.z = tile[(c + 2) * TPAD + r];
  o.w = tile[(c + 3) * TPAD + r];
  *(v4f*)(&out[(bx * TILE + r) * LC + by * TILE + c]) = o;
}

// Generic fallback (workspace too small or L != 4096): strided column reads.
__global__ void PNet_direct_kernel(const int* __restrict__ w,
                                   const float* __restrict__ z,
                                   const float* __restrict__ Wm,
                                   const float* __restrict__ b,
                                   float* __restrict__ out, int L) {
  const int i  = blockIdx.x;
  const int wi = w[i];
  const float zw = z[(size_t)i * L + wi];
  for (int j = threadIdx.x; j < L; j += blockDim.x)
    out[(size_t)i * L + j] = fmaf(zw, Wm[(size_t)j * L + wi], b[j]);
}

extern "C" void kernel_launch(void* const* d_in, const int* in_sizes, int n_in,
                              void* d_out, int out_size, void* d_ws, size_t ws_size,
                              hipStream_t stream) {
  const int*   w  = (const int*)d_in[0];
  const float* z  = (const float*)d_in[1];
  const float* Wm = (const float*)d_in[2];
  const float* b  = (const float*)d_in[3];
  float* out = (float*)d_out;

  const int N = in_sizes[0];   // 16384
  const int L = in_sizes[3];   // 4096

  const bool fast = (L == LC) &&
                    (ws_size >= (size_t)LC * LC * sizeof(float));

  if (fast) {
    float* Wt = (float*)d_ws;
    dim3 tg(LC / TILE, LC / TILE);
    PNet_transpose_kernel<<<tg, THREADS, 0, stream>>>(Wm, Wt);
    PNet_row_kernel<<<N, THREADS, 0, stream>>>(w, z, Wt, b, out);
  } else {
    PNet_direct_kernel<<<N, 256, 0, stream>>>(w, z, Wm, b, out, L);
  }
}